// Sinkhron_14285061227130
// MI455X (gfx1250) — compile-verified
//
#include <hip/hip_runtime.h>
#include <hip/hip_bf16.h>

#define NN    512          // N (matches reference setup)
#define ITER  50
#define INVB  2.0f         // 1/beta, beta = 0.5
#define EPSF  1e-8f
#define TOLF  0.1f

typedef __attribute__((ext_vector_type(2))) float v2f;
typedef __attribute__((ext_vector_type(8))) float v8f;

// ---------------------------------------------------------------- setup ----
__global__ void k_setup(const int* __restrict__ mask,
                        float* msum, float* D, float* S, float* sig,
                        float* del, float* sgp,
                        float* errArr, int* doneArr, int* keffArr,
                        float* dist, int bs) {
    int b = blockIdx.x, t = threadIdx.x;
    __shared__ float red[256];
    float c = 0.f;
    for (int i = t; i < NN; i += 256) c += (mask[b*NN + i] != 0) ? 1.f : 0.f;
    red[t] = c; __syncthreads();
    for (int s = 128; s > 0; s >>= 1) { if (t < s) red[t] += red[t + s]; __syncthreads(); }
    float ms = red[0];
    if (t == 0) msum[b] = ms;
    float inv = 1.f / ms;
    for (int i = t; i < NN; i += 256) {
        int gi = b*NN + i;
        bool m = mask[gi] != 0;
        D[gi] = 1.f; S[gi] = 1.f;
        sig[gi] = m ? inv : 0.f;     // sigma0 = mu
        del[gi] = 0.f; sgp[gi] = 0.f;
    }
    if (b == 0) {
        for (int i = t; i <= ITER; i += 256) { errArr[i] = 0.f; doneArr[i] = 0; keffArr[i] = 0; }
        for (int i = t; i < bs; i += 256) dist[i] = 0.f;
    }
}

// ------------------------------------------------------- pass1: b = Q*sigma
// b_i = m_i D_i * sum_j E_{k+1}[i,j] * (m_j S_j sigma_j);   delta = mu / b
__global__ void k_pass1(const float* __restrict__ C, const int* __restrict__ mask,
                        const float* __restrict__ D, const float* __restrict__ S,
                        const float* __restrict__ sig, const float* __restrict__ msum,
                        float* __restrict__ del, const int* __restrict__ keffArr, int k) {
    int strip = blockIdx.x, b = blockIdx.y, t = threadIdx.x;
    __shared__ float tile[16][NN];
    __shared__ float w[NN];
    __shared__ float bacc[16];
    float kk1 = (float)(keffArr[k - 1] + 1);
    float sc  = -kk1 * INVB;                        // exp(sc * c) = exp(-(k+1)c/beta)
    const float* Crow = C + ((size_t)b * NN + (size_t)strip * 16) * NN;
    for (int idx = t; idx < 16 * NN; idx += 256) {
        int rr = idx / NN, cc = idx % NN;
        tile[rr][cc] = Crow[(size_t)rr * NN + cc];
    }
    // warm L2 for the next strip during the cold pass (speculative prefetch)
    if (strip + 1 < NN / 16) {
        const float* nxt = Crow + (size_t)16 * NN;
        for (int idx = t * 16; idx < 16 * NN; idx += 256 * 16)
            __builtin_prefetch(&nxt[idx], 0, 1);
    }
    for (int j = t; j < NN; j += 256) {
        int gj = b*NN + j;
        w[j] = (mask[gj] != 0) ? S[gj] * sig[gj] : 0.f;
    }
    if (t < 16) bacc[t] = 0.f;
    __syncthreads();

    int wv = t >> 5, lane = t & 31;
    int r = lane & 15, h = lane >> 4;
    float selB = (r == 0) ? 1.f : 0.f;              // arithmetic mask: no EXEC juggling
    v8f acc = {0.f,0.f,0.f,0.f,0.f,0.f,0.f,0.f};
    for (int jt = wv; jt < NN/16; jt += 8) {        // uniform per wave
        int j0 = jt * 16;
        #pragma unroll
        for (int s4 = 0; s4 < 4; ++s4) {
            int jb = j0 + s4 * 4;
            // A operand: E-tile, lane r holds K = 2h, 2h+1
            v2f a; a.x = __expf(sc * tile[r][jb + 2*h]);
                   a.y = __expf(sc * tile[r][jb + 2*h + 1]);
            // B operand: weights in column 0 only -> D[:,0] = E @ w
            v2f bb; bb.x = selB * w[jb + 2*h];
                    bb.y = selB * w[jb + 2*h + 1];
            acc = __builtin_amdgcn_wmma_f32_16x16x4_f32(
                      false, a, false, bb, (short)0, acc, false, false);
        }
    }
    if (r == 0) {                 // column 0 lives in lanes 0 (M=0..7) and 16 (M=8..15)
        #pragma unroll
        for (int v = 0; v < 8; ++v) atomicAdd(&bacc[h*8 + v], acc[v]);
    }
    __syncthreads();
    if (t < 16) {
        int gi = b*NN + strip*16 + t;
        bool m = mask[gi] != 0;
        float bval = m ? D[gi] * bacc[t] : EPSF;   // masked_fill(EPS)
        float mu   = m ? (1.f / msum[b]) : 0.f;
        del[gi] = mu / bval;
    }
}

// ------------------------------------------------------ pass2: a = Q^T*delta
// a_j = m_j S_j * sum_i E_{k+1}[i,j] * (m_i D_i delta_i);   sigma' = mv / a
__global__ void k_pass2(const float* __restrict__ C, const int* __restrict__ mask,
                        const float* __restrict__ D, const float* __restrict__ S,
                        const float* __restrict__ del, const float* __restrict__ msum,
                        float* __restrict__ sgp, const int* __restrict__ keffArr, int k) {
    int cs = blockIdx.x, b = blockIdx.y, t = threadIdx.x;
    __shared__ float tile[NN][17];     // padded: avoid bank conflicts
    __shared__ float dv[NN];
    __shared__ float aacc[16];
    float kk1 = (float)(keffArr[k - 1] + 1);
    float sc  = -kk1 * INVB;
    const float* Cb = C + (size_t)b * NN * NN + (size_t)cs * 16;
    for (int idx = t; idx < NN * 16; idx += 256) {
        int rr = idx >> 4, cc = idx & 15;
        tile[rr][cc] = Cb[(size_t)rr * NN + cc];
    }
    for (int i = t; i < NN; i += 256) {
        int gi = b*NN + i;
        dv[i] = (mask[gi] != 0) ? D[gi] * del[gi] : 0.f;
    }
    if (t < 16) aacc[t] = 0.f;
    __syncthreads();

    int wv = t >> 5, lane = t & 31;
    int r = lane & 15, h = lane >> 4;
    float selA = (r == 0) ? 1.f : 0.f;              // arithmetic mask
    v8f acc = {0.f,0.f,0.f,0.f,0.f,0.f,0.f,0.f};
    #pragma unroll 4
    for (int sl = 0; sl < 16; ++sl) {               // rows [wv*64, wv*64+64)
        int ib = wv * 64 + sl * 4;
        // A operand: delta-weights in row 0 only -> D[0,:] = d^T @ E
        v2f a; a.x = selA * dv[ib + 2*h];
               a.y = selA * dv[ib + 2*h + 1];
        // B operand: E slab, lane holds rows K = 2h, 2h+1 at column r
        v2f bb; bb.x = __expf(sc * tile[ib + 2*h    ][r]);
                bb.y = __expf(sc * tile[ib + 2*h + 1][r]);
        acc = __builtin_amdgcn_wmma_f32_16x16x4_f32(
                  false, a, false, bb, (short)0, acc, false, false);
    }
    if (h == 0) atomicAdd(&aacc[r], acc[0]);   // D[0][n]: VGPR0, lanes 0..15
    __syncthreads();
    if (t < 16) {
        int gj = b*NN + cs*16 + t;
        bool m = mask[gj] != 0;
        float aval = m ? S[gj] * aacc[t] : EPSF;
        float mv   = m ? (1.f / msum[b]) : 0.f;
        sgp[gj] = mv / aval;
    }
}

// --------------------------------------------------------------- pass3: err
// err_i = m_i D_i * sum_j m_j S_j * | delta_i sigma'_j E_{k+1} - E_k |
__global__ void k_pass3(const float* __restrict__ C, const int* __restrict__ mask,
                        const float* __restrict__ D, const float* __restrict__ S,
                        const float* __restrict__ del, const float* __restrict__ sgp,
                        float* __restrict__ errArr, const int* __restrict__ keffArr, int k) {
    int strip = blockIdx.x, b = blockIdx.y, t = threadIdx.x;
    __shared__ float tile[16][NN];
    __shared__ float wS[NN], wsp[NN];
    __shared__ float red[256];
    float kf  = (float)keffArr[k - 1];
    float sck = -kf * INVB;                        // E_k   = exp(sck * c)
    const float* Crow = C + ((size_t)b * NN + (size_t)strip * 16) * NN;
    for (int idx = t; idx < 16 * NN; idx += 256) {
        int rr = idx / NN, cc = idx % NN;
        tile[rr][cc] = Crow[(size_t)rr * NN + cc];
    }
    for (int j = t; j < NN; j += 256) {
        int gj = b*NN + j;
        wS[j]  = (mask[gj] != 0) ? S[gj] : 0.f;
        wsp[j] = sgp[gj];
    }
    __syncthreads();
    int row = t >> 4, sub = t & 15;
    int gi = b*NN + strip*16 + row;
    float dl = del[gi];
    float Di = (mask[gi] != 0) ? D[gi] : 0.f;
    float ssum = 0.f;
    for (int j = sub; j < NN; j += 16) {
        float c   = tile[row][j];
        float ek  = __expf(sck * c);
        float ek1 = ek * __expf(-INVB * c);        // E_{k+1} = E_k * exp(-c/beta)
        ssum += wS[j] * fabsf(dl * wsp[j] * ek1 - ek);
    }
    red[t] = ssum; __syncthreads();
    for (int s = 8; s > 0; s >>= 1) {
        if (sub < s) red[t] += red[t + s];
        __syncthreads();
    }
    if (sub == 0) {
        float err_i = Di * red[row * 16];          // >= 0, so float-bit max is valid
        atomicMax((unsigned int*)&errArr[k], __float_as_uint(err_i));
    }
}

// ------------------------------------------------------------------ control
__global__ void k_ctrl(float* D, float* S, float* sig,
                       const float* __restrict__ del, const float* __restrict__ sgp,
                       const float* __restrict__ errArr, int* doneArr, int* keffArr, int k) {
    int b = blockIdx.x, t = threadIdx.x;
    int done_prev = doneArr[k - 1];
    if (!done_prev) {                           // update gated by done BEFORE this iter's OR
        for (int i = t; i < NN; i += 256) {
            int gi = b*NN + i;
            D[gi] *= del[gi];
            S[gi] *= sgp[gi];
            sig[gi] = sgp[gi];
        }
    }
    if (b == 0 && t == 0) {
        float e = errArr[k];
        doneArr[k] = done_prev | (e < TOLF ? 1 : 0);
        keffArr[k] = keffArr[k - 1] + (done_prev ? 0 : 1);
    }
}

// ----------------------------------------------------------------- finalize
__global__ void k_final(const float* __restrict__ C, const int* __restrict__ mask,
                        const float* __restrict__ D, const float* __restrict__ S,
                        const int* __restrict__ keffArr,
                        float* __restrict__ dist, float* __restrict__ Tout,
                        float* __restrict__ Cout) {
    int strip = blockIdx.x, b = blockIdx.y, t = threadIdx.x;
    __shared__ float wS[NN];
    __shared__ float Drow[16];
    __shared__ float red[256];
    float kf = (float)keffArr[ITER];
    float sc = -kf * INVB;
    for (int j = t; j < NN; j += 256) {
        int gj = b*NN + j;
        wS[j] = (mask[gj] != 0) ? S[gj] : 0.f;
    }
    if (t < 16) {
        int gi = b*NN + strip*16 + t;
        Drow[t] = (mask[gi] != 0) ? D[gi] : 0.f;
    }
    __syncthreads();
    float lsum = 0.f;
    size_t base = ((size_t)b * NN + (size_t)strip * 16) * NN;
    for (int idx = t; idx < 16 * NN; idx += 256) {
        int rr = idx / NN, cc = idx % NN;
        size_t g = base + (size_t)rr * NN + cc;
        float c  = C[g];
        float Tv = Drow[rr] * wS[cc] * __expf(sc * c);
        Tout[g] = Tv;
        Cout[g] = c;
        lsum += Tv * c;
    }
    red[t] = lsum; __syncthreads();
    for (int s = 128; s > 0; s >>= 1) { if (t < s) red[t] += red[t + s]; __syncthreads(); }
    if (t == 0) atomicAdd(&dist[b], red[0]);
}

// ------------------------------------------------------------------- launch
extern "C" void kernel_launch(void* const* d_in, const int* in_sizes, int n_in,
                              void* d_out, int out_size, void* d_ws, size_t ws_size,
                              hipStream_t stream) {
    const float* C  = (const float*)d_in[0];
    const int* mask = (const int*)d_in[4];

    int bsN = in_sizes[4];                 // bs * N
    int N   = in_sizes[0] / bsN;           // = 512
    int bs  = bsN / N;                     // = 128
    (void)N;                               // kernels assume N == NN (512)

    size_t bn = (size_t)bs * NN;
    float* wsf    = (float*)d_ws;
    float* D      = wsf;
    float* S      = D + bn;
    float* sig    = S + bn;
    float* del    = sig + bn;
    float* sgp    = del + bn;
    float* msum   = sgp + bn;
    float* errArr = msum + bs;
    int*   doneArr = (int*)(errArr + (ITER + 1));
    int*   keffArr = doneArr + (ITER + 1);

    float* dist = (float*)d_out;
    float* Tout = dist + bs;
    float* Cout = Tout + (size_t)bs * NN * NN;

    dim3 gs(NN / 16, bs);
    k_setup<<<bs, 256, 0, stream>>>(mask, msum, D, S, sig, del, sgp,
                                    errArr, doneArr, keffArr, dist, bs);
    for (int k = 1; k <= ITER; ++k) {
        k_pass1<<<gs, 256, 0, stream>>>(C, mask, D, S, sig, msum, del, keffArr, k);
        k_pass2<<<gs, 256, 0, stream>>>(C, mask, D, S, del, msum, sgp, keffArr, k);
        k_pass3<<<gs, 256, 0, stream>>>(C, mask, D, S, del, sgp, errArr, keffArr, k);
        k_ctrl <<<bs, 256, 0, stream>>>(D, S, sig, del, sgp, errArr, doneArr, keffArr, k);
    }
    k_final<<<gs, 256, 0, stream>>>(C, mask, D, S, keffArr, dist, Tout, Cout);
}